// GraphConvBlock_81990925680656
// MI455X (gfx1250) — compile-verified
//
#include <hip/hip_runtime.h>
#include <math.h>

typedef float v2f __attribute__((ext_vector_type(2)));
typedef float v8f __attribute__((ext_vector_type(8)));

#define DIM 128
#define LN_EPS 1e-5f
#define WSTR 33   // padded LDS row stride (floats) -> 16 lanes hit 16 distinct banks

// ---- order-preserving float <-> uint key (for integer atomicMax on floats) ----
__device__ __forceinline__ unsigned f2k(float f) {
    unsigned u = __float_as_uint(f);
    return (u & 0x80000000u) ? ~u : (u | 0x80000000u);
}
__device__ __forceinline__ float k2f(unsigned k) {
    return __uint_as_float((k & 0x80000000u) ? (k ^ 0x80000000u) : ~k);
}
// enc(-inf) = ~0xFF800000 = 0x007FFFFF
#define NEG_INF_KEY 0x007FFFFFu

__device__ __forceinline__ float gelu_exact(float v) {
    return 0.5f * v * (1.0f + erff(v * 0.70710678118654752440f));
}

// ---- CDNA5 async global -> LDS staging (ASYNCcnt-tracked) ----
// Stage one 32-float weight row segment (128 B) per lane iteration; 8 x b128.
__device__ __forceinline__ void stage_row_async(float* ldsrow, const float* grow) {
    unsigned l0 = (unsigned)(uintptr_t)ldsrow;   // low 32 bits = LDS byte offset
    #pragma unroll
    for (int i = 0; i < 8; ++i) {
        asm volatile("global_load_async_to_lds_b128 %0, %1, off"
                     :: "v"(l0 + 32u * i), "v"(grow + 8 * i)
                     : "memory");
    }
}
__device__ __forceinline__ void wait_async_all() {
    asm volatile("s_wait_asynccnt 0" ::: "memory");
}

// ---------------- kernel 1: init agg keys to enc(-inf) ----------------
__global__ void __launch_bounds__(256) init_agg_kernel(unsigned* __restrict__ aggk, int total) {
    int i = blockIdx.x * 256 + threadIdx.x;
    if (i < total) aggk[i] = NEG_INF_KEY;
}

// ---------------- kernel 2: edge scatter-max (1 wave per edge) ----------------
__global__ void __launch_bounds__(256) scatter_kernel(
    const float* __restrict__ x, const long long* __restrict__ ei,
    const float* __restrict__ ew, unsigned* __restrict__ aggk, int E)
{
    const int e = blockIdx.x * 8 + (threadIdx.x >> 5);
    if (e >= E) return;
    const int lane = threadIdx.x & 31;
    const int src = (int)ei[e];
    const int dst = (int)ei[(size_t)E + e];
    const float w = ew[e];
    const float4 xv = *((const float4*)(x + (size_t)src * DIM) + lane);
    unsigned* ap = aggk + (size_t)dst * DIM + lane * 4;
    atomicMax(ap + 0, f2k(xv.x * w));
    atomicMax(ap + 1, f2k(xv.y * w));
    atomicMax(ap + 2, f2k(xv.z * w));
    atomicMax(ap + 3, f2k(xv.w * w));
}

// ---------------- kernel 3: h = LN(gelu(agg@Wrel^T + brel + x@Wroot^T) + x) ----------------
// Virtual K = 256: chunks 0..3 from (agg, Wrel), chunks 4..7 from (x, Wroot).
// Weight chunks (128 cols x 32 K) double-buffered in LDS via async copies.
__global__ void __launch_bounds__(128) gemm1_kernel(
    const unsigned* __restrict__ aggk, const float* __restrict__ x,
    const float* __restrict__ Wrel, const float* __restrict__ brel,
    const float* __restrict__ Wroot, const float* __restrict__ gamma,
    const float* __restrict__ beta, float* __restrict__ h, int N)
{
    __shared__ __align__(16) float wlds[2][128 * WSTR];

    const int tid   = threadIdx.x;
    const int lane  = tid & 31;
    const int wave  = tid >> 5;
    const int row0  = (blockIdx.x * 4 + wave) * 16;
    const int l15   = lane & 15;
    const int khalf = (lane >> 4) * 2;           // K sub-offset per lane half
    const int rowA  = min(row0 + l15, N - 1);    // clamp: keep EXEC all-1s

    v8f c[8];
    #pragma unroll
    for (int j = 0; j < 8; ++j) c[j] = (v8f){0.f,0.f,0.f,0.f,0.f,0.f,0.f,0.f};

    const unsigned* ap = aggk + (size_t)rowA * DIM;
    const float*    xp = x    + (size_t)rowA * DIM;

    // prologue: stage chunk 0 (Wrel cols, K 0..31)
    stage_row_async(&wlds[0][tid * WSTR], Wrel + (size_t)tid * DIM);
    wait_async_all();
    __syncthreads();

    for (int ch = 0; ch < 8; ++ch) {
        if (ch < 7) {   // prefetch next chunk into the other buffer
            const int nc = ch + 1;
            const float* Wsrc = (nc < 4) ? (Wrel + nc * 32)
                                         : (Wroot + (size_t)(nc - 4) * 32);
            stage_row_async(&wlds[nc & 1][tid * WSTR], Wsrc + (size_t)tid * DIM);
        }
        const float* lb = wlds[ch & 1];
        for (int k0 = 0; k0 < 32; k0 += 4) {
            const int kl = k0 + khalf;
            v2f a;
            if (ch < 4) {               // A = agg (decode keys, -inf -> 0)
                const int kg = ch * 32 + kl;
                float a0 = k2f(ap[kg]);     a[0] = (a0 == -INFINITY) ? 0.0f : a0;
                float a1 = k2f(ap[kg + 1]); a[1] = (a1 == -INFINITY) ? 0.0f : a1;
            } else {                    // A = x
                const int kg = (ch - 4) * 32 + kl;
                a[0] = xp[kg]; a[1] = xp[kg + 1];
            }
            #pragma unroll
            for (int j = 0; j < 8; ++j) {
                const float* bp = lb + (16 * j + l15) * WSTR + kl;
                v2f b; b[0] = bp[0]; b[1] = bp[1];
                c[j] = __builtin_amdgcn_wmma_f32_16x16x4_f32(false, a, false, b,
                                                             (short)0, c[j], false, false);
            }
        }
        if (ch < 7) { wait_async_all(); __syncthreads(); }
    }

    // epilogue: + bias, gelu, + skip(x)    (C/D layout: m = v + 8*(lane>=16), n = 16j + l15)
    const int mhalf = (lane >> 4) * 8;
    #pragma unroll
    for (int j = 0; j < 8; ++j) {
        const int ng = 16 * j + l15;
        const float bj = brel[ng];
        #pragma unroll
        for (int v = 0; v < 8; ++v) {
            const int rg = min(row0 + mhalf + v, N - 1);
            c[j][v] = gelu_exact(c[j][v] + bj) + x[(size_t)rg * DIM + ng];
        }
    }
    // LayerNorm: rows are spread over 16 lanes (same half) x 8 j-tiles
    float s[8], q[8];
    #pragma unroll
    for (int v = 0; v < 8; ++v) { s[v] = 0.f; q[v] = 0.f; }
    #pragma unroll
    for (int j = 0; j < 8; ++j) {
        #pragma unroll
        for (int v = 0; v < 8; ++v) { float t = c[j][v]; s[v] += t; q[v] += t * t; }
    }
    #pragma unroll
    for (int m = 1; m <= 8; m <<= 1) {
        #pragma unroll
        for (int v = 0; v < 8; ++v) {
            s[v] += __shfl_xor(s[v], m, 32);
            q[v] += __shfl_xor(q[v], m, 32);
        }
    }
    float mu[8], rs[8];
    #pragma unroll
    for (int v = 0; v < 8; ++v) {
        mu[v] = s[v] * (1.0f / 128.0f);
        float var = q[v] * (1.0f / 128.0f) - mu[v] * mu[v];
        rs[v] = rsqrtf(var + LN_EPS);
    }
    #pragma unroll
    for (int j = 0; j < 8; ++j) {
        const int ng = 16 * j + l15;
        const float g = gamma[ng], bt = beta[ng];
        #pragma unroll
        for (int v = 0; v < 8; ++v) {
            const int rg = row0 + mhalf + v;
            if (rg < N)
                h[(size_t)rg * DIM + ng] = (c[j][v] - mu[v]) * rs[v] * g + bt;
        }
    }
}

// ---------------- kernel 4: h = LN(gelu(h@linW^T + linb) + h), in place ----------------
__global__ void __launch_bounds__(128) gemm2_kernel(
    const float* __restrict__ Wl, const float* __restrict__ bl,
    const float* __restrict__ gamma, const float* __restrict__ beta,
    float* h, int N)
{
    __shared__ __align__(16) float wlds[2][128 * WSTR];

    const int tid   = threadIdx.x;
    const int lane  = tid & 31;
    const int wave  = tid >> 5;
    const int row0  = (blockIdx.x * 4 + wave) * 16;
    const int l15   = lane & 15;
    const int khalf = (lane >> 4) * 2;
    const int rowA  = min(row0 + l15, N - 1);

    v8f c[8];
    #pragma unroll
    for (int j = 0; j < 8; ++j) c[j] = (v8f){0.f,0.f,0.f,0.f,0.f,0.f,0.f,0.f};

    const float* hp = h + (size_t)rowA * DIM;

    stage_row_async(&wlds[0][tid * WSTR], Wl + (size_t)tid * DIM);
    wait_async_all();
    __syncthreads();

    for (int ch = 0; ch < 4; ++ch) {
        if (ch < 3) {
            const int nc = ch + 1;
            stage_row_async(&wlds[nc & 1][tid * WSTR], Wl + nc * 32 + (size_t)tid * DIM);
        }
        const float* lb = wlds[ch & 1];
        for (int k0 = 0; k0 < 32; k0 += 4) {
            const int kl = k0 + khalf;
            const int kg = ch * 32 + kl;
            v2f a; a[0] = hp[kg]; a[1] = hp[kg + 1];
            #pragma unroll
            for (int j = 0; j < 8; ++j) {
                const float* bp = lb + (16 * j + l15) * WSTR + kl;
                v2f b; b[0] = bp[0]; b[1] = bp[1];
                c[j] = __builtin_amdgcn_wmma_f32_16x16x4_f32(false, a, false, b,
                                                             (short)0, c[j], false, false);
            }
        }
        if (ch < 3) { wait_async_all(); __syncthreads(); }
    }

    const int mhalf = (lane >> 4) * 8;
    #pragma unroll
    for (int j = 0; j < 8; ++j) {
        const int ng = 16 * j + l15;
        const float bj = bl[ng];
        #pragma unroll
        for (int v = 0; v < 8; ++v) {
            const int rg = min(row0 + mhalf + v, N - 1);
            c[j][v] = gelu_exact(c[j][v] + bj) + h[(size_t)rg * DIM + ng];
        }
    }
    float s[8], q[8];
    #pragma unroll
    for (int v = 0; v < 8; ++v) { s[v] = 0.f; q[v] = 0.f; }
    #pragma unroll
    for (int j = 0; j < 8; ++j) {
        #pragma unroll
        for (int v = 0; v < 8; ++v) { float t = c[j][v]; s[v] += t; q[v] += t * t; }
    }
    #pragma unroll
    for (int m = 1; m <= 8; m <<= 1) {
        #pragma unroll
        for (int v = 0; v < 8; ++v) {
            s[v] += __shfl_xor(s[v], m, 32);
            q[v] += __shfl_xor(q[v], m, 32);
        }
    }
    float mu[8], rs[8];
    #pragma unroll
    for (int v = 0; v < 8; ++v) {
        mu[v] = s[v] * (1.0f / 128.0f);
        float var = q[v] * (1.0f / 128.0f) - mu[v] * mu[v];
        rs[v] = rsqrtf(var + LN_EPS);
    }
    #pragma unroll
    for (int j = 0; j < 8; ++j) {
        const int ng = 16 * j + l15;
        const float g = gamma[ng], bt = beta[ng];
        #pragma unroll
        for (int v = 0; v < 8; ++v) {
            const int rg = row0 + mhalf + v;
            if (rg < N)
                h[(size_t)rg * DIM + ng] = (c[j][v] - mu[v]) * rs[v] * g + bt;
        }
    }
}

extern "C" void kernel_launch(void* const* d_in, const int* in_sizes, int n_in,
                              void* d_out, int out_size, void* d_ws, size_t ws_size,
                              hipStream_t stream) {
    const float*     x     = (const float*)d_in[0];
    const long long* ei    = (const long long*)d_in[1];
    const float*     ew    = (const float*)d_in[2];
    const float*     Wrel  = (const float*)d_in[3];
    const float*     brel  = (const float*)d_in[4];
    const float*     Wroot = (const float*)d_in[5];
    const float*     linW  = (const float*)d_in[6];
    const float*     linb  = (const float*)d_in[7];
    const float*     gamma = (const float*)d_in[8];
    const float*     beta  = (const float*)d_in[9];

    const int N = in_sizes[0] / DIM;
    const int E = in_sizes[2];
    float*    out  = (float*)d_out;
    unsigned* aggk = (unsigned*)d_ws;   // N*DIM u32 keys (10.24 MB)

    const int nd = N * DIM;
    init_agg_kernel<<<(nd + 255) / 256, 256, 0, stream>>>(aggk, nd);
    scatter_kernel<<<(E + 7) / 8, 256, 0, stream>>>(x, ei, ew, aggk, E);

    const int gblocks = (N + 63) / 64;   // 4 waves/block, 16 rows/wave
    gemm1_kernel<<<gblocks, 128, 0, stream>>>(aggk, x, Wrel, brel, Wroot,
                                              gamma, beta, out, N);
    gemm2_kernel<<<gblocks, 128, 0, stream>>>(linW, linb, gamma, beta, out, N);

    hipMemcpyAsync(out + (size_t)nd, ew, (size_t)E * sizeof(float),
                   hipMemcpyDeviceToDevice, stream);
}